// RotaryPositionalEmbedding_12386685681708
// MI455X (gfx1250) — compile-verified
//
#include <hip/hip_runtime.h>

// RoPE, fp32, [BATCH=32, SEQ=4096, D_K=128].
// Memory-bound streaming kernel: 128 MiB total traffic, fits in MI455X's
// 192 MB L2, so default temporal hints are used (keep data L2-resident).
// Trig is computed once per (position, pair) and reused across all 32 batch
// rows (they share identical angles), so accurate sincosf is ~free.

constexpr int kBatch      = 32;
constexpr int kSeq        = 4096;
constexpr int kD          = 128;
constexpr int kVecPerRow  = kD / 4;          // 32 float4 per (b,p) row
constexpr int kPosPerBlock = 8;              // 8 positions/block
constexpr int kThreads    = kPosPerBlock * kVecPerRow;  // 256 threads (8 waves)

__global__ __launch_bounds__(kThreads)
void rope_f32_kernel(const float* __restrict__ x,
                     const int*   __restrict__ tok_pos,
                     float*       __restrict__ out) {
    const int tid  = threadIdx.x;
    const int quad = tid & (kVecPerRow - 1);                 // which float4 in the row: 0..31
    const int p    = blockIdx.x * kPosPerBlock + (tid >> 5); // sequence position: 0..4095

    const float fpos = (float)tok_pos[p];

    // Pair indices handled by this lane: j0 = 2*quad, j1 = 2*quad+1.
    // exponent_j = 2*j / d ;  inv_freq = theta^(-exponent) = exp2(-exponent*log2(theta))
    const float log2_theta = 13.287712379549449f;            // log2(10000)
    const float inv_d      = 1.0f / (float)kD;
    const float e0 = (float)(4 * quad)     * inv_d;          // 2*(2q)/128
    const float e1 = (float)(4 * quad + 2) * inv_d;          // 2*(2q+1)/128
    const float a0 = fpos * exp2f(-e0 * log2_theta);
    const float a1 = fpos * exp2f(-e1 * log2_theta);

    // Accurate range-reduced sin/cos (angles up to ~4095 rad); cost amortized
    // over the 32-batch reuse loop below.
    float s0, c0, s1, c1;
    sincosf(a0, &s0, &c0);
    sincosf(a1, &s1, &c1);

    const size_t row     = (size_t)p * kVecPerRow + quad;    // float4 index within one batch
    const size_t bstride = (size_t)kSeq * kVecPerRow;        // float4 stride between batches

    const float4* __restrict__ xv = (const float4*)x;
    float4*       __restrict__ ov = (float4*)out;

#pragma unroll 8
    for (int b = 0; b < kBatch; ++b) {
        const size_t idx = row + (size_t)b * bstride;
        float4 v = xv[idx];          // global_load_b128, fully coalesced per wave32
        float4 r;
        r.x = v.x * c0 - v.y * s0;   // even' = e*cos - o*sin
        r.y = v.x * s0 + v.y * c0;   // odd'  = e*sin + o*cos
        r.z = v.z * c1 - v.w * s1;
        r.w = v.z * s1 + v.w * c1;
        ov[idx] = r;                 // global_store_b128
    }
}

extern "C" void kernel_launch(void* const* d_in, const int* in_sizes, int n_in,
                              void* d_out, int out_size, void* d_ws, size_t ws_size,
                              hipStream_t stream) {
    (void)in_sizes; (void)n_in; (void)d_ws; (void)ws_size; (void)out_size;
    const float* x   = (const float*)d_in[0];
    const int*   pos = (const int*)d_in[1];
    float*       out = (float*)d_out;

    dim3 grid(kSeq / kPosPerBlock);   // 512 blocks
    dim3 block(kThreads);             // 256 threads = 8 wave32
    rope_f32_kernel<<<grid, block, 0, stream>>>(x, pos, out);
}